// Renderer_17403207483410
// MI455X (gfx1250) — compile-verified
//
#include <hip/hip_runtime.h>
#include <math.h>

// MERL BRDF renderer for MI455X (gfx1250, wave32).
//   Kernel 1: repack BRDF table [3][90][90][180] -> [90*90*180] float4 (RGB + pad),
//             per-channel K constants folded in. Lives in d_ws, stays L2-resident.
//   Kernel 2: per-ray shading. Streams AoS inputs via async global->LDS b128 copies
//             (CDNA5 ASYNCcnt path), gathers 8 float4 corners, trilinear interp.

#define KTILE 256
#define TH_N 90
#define TD_N 90
#define PD_N 180
#define PI_F 3.14159265358979323846f
#define RAD2IDX 57.295779513082323f /* 180/pi == 90/(pi/2) */

#if __has_builtin(__builtin_amdgcn_global_load_async_to_lds_b128)
#define HAVE_ASYNC_LDS 1
#else
#define HAVE_ASYNC_LDS 0
#endif

#if HAVE_ASYNC_LDS
typedef int v4i_t __attribute__((vector_size(16)));
typedef __attribute__((address_space(1))) v4i_t* gptr_v4i;
typedef __attribute__((address_space(3))) v4i_t* lptr_v4i;
#endif

__device__ __forceinline__ void norm3(float& x, float& y, float& z) {
    // matches reference: v / (sqrt(dot(v,v)) + 1e-12)
    float s = sqrtf(x * x + y * y + z * z) + 1e-12f;
    float r = 1.0f / s;
    x *= r; y *= r; z *= r;
}

__global__ __launch_bounds__(KTILE) void permute_brdf_kernel(
    const float* __restrict__ brdf, float4* __restrict__ tbl, int texels) {
    int t = blockIdx.x * blockDim.x + threadIdx.x;
    if (t >= texels) return;
    float4 v;
    v.x = brdf[t]              * (1.00f / 1500.0f);
    v.y = brdf[t + texels]     * (1.15f / 1500.0f);
    v.z = brdf[t + 2 * texels] * (1.66f / 1500.0f);
    v.w = 0.0f;
    tbl[t] = v;
}

template <bool USE_WS>
__global__ __launch_bounds__(KTILE) void render_kernel(
    const float* __restrict__ brdf, const float4* __restrict__ tbl,
    const float* __restrict__ normal, const float* __restrict__ view,
    const float* __restrict__ light, const float* __restrict__ inten,
    float* __restrict__ out, int n) {
    __shared__ __align__(16) float s_in[4 * 3 * KTILE];  // 12 KB: n,v,l,intensity tiles
    const int t = threadIdx.x;
    const int tile0 = blockIdx.x * KTILE;
    const bool full = (tile0 + KTILE) <= n;  // uniform per block

    if (full) {
        const float* srcs[4] = {normal, view, light, inten};
        // 768 x 16B chunks (4 arrays x 192 chunks); 3 chunks per thread.
#pragma unroll
        for (int k = 0; k < 3; ++k) {
            int c = t + k * KTILE;
            int arr = c / 192;
            int off = (c - arr * 192) * 4;  // float offset within 768-float region
            const float* src = srcs[arr] + (size_t)tile0 * 3 + off;
            float* dst = s_in + arr * (3 * KTILE) + off;
#if HAVE_ASYNC_LDS
            __builtin_amdgcn_global_load_async_to_lds_b128(
                (gptr_v4i)src, (lptr_v4i)dst, 0, 0);
#else
            *(float4*)dst = *(const float4*)src;
#endif
        }
#if HAVE_ASYNC_LDS
#if __has_builtin(__builtin_amdgcn_s_wait_asynccnt)
        __builtin_amdgcn_s_wait_asynccnt(0);
#else
        asm volatile("s_wait_asynccnt 0" ::: "memory");
#endif
#endif
        __syncthreads();
    }

    const int i = tile0 + t;
    if (i >= n) return;

    float nx, ny, nz, vx, vy, vz, lx, ly, lz, ir, ig, ibl;
    if (full) {
        int b = t * 3;
        nx = s_in[b];          ny = s_in[b + 1];          nz = s_in[b + 2];
        vx = s_in[768 + b];    vy = s_in[768 + b + 1];    vz = s_in[768 + b + 2];
        lx = s_in[1536 + b];   ly = s_in[1536 + b + 1];   lz = s_in[1536 + b + 2];
        ir = s_in[2304 + b];   ig = s_in[2304 + b + 1];   ibl = s_in[2304 + b + 2];
    } else {
        size_t b = (size_t)i * 3;
        nx = normal[b]; ny = normal[b + 1]; nz = normal[b + 2];
        vx = view[b];   vy = view[b + 1];   vz = view[b + 2];
        lx = light[b];  ly = light[b + 1];  lz = light[b + 2];
        ir = inten[b];  ig = inten[b + 1];  ibl = inten[b + 2];
    }

    norm3(nx, ny, nz);
    norm3(vx, vy, vz);
    norm3(lx, ly, lz);

    // half vector
    float hx = lx + vx, hy = ly + vy, hz = lz + vz;
    norm3(hx, hy, hz);

    float hn = hx * nx + hy * ny + hz * nz;
    float hl = hx * lx + hy * ly + hz * lz;
    float th = acosf(fminf(fmaxf(hn, -1.0f), 1.0f));
    float td = acosf(fminf(fmaxf(hl, -1.0f), 1.0f));

    // tangent a = -n + (n.h) h ; degenerate guard
    float ax = hn * hx - nx, ay = hn * hy - ny, az = hn * hz - nz;
    float aa = ax * ax + ay * ay + az * az;
    bool degen = (aa <= 1.0e-6f);
    if (degen) { ax = 1.0f; ay = 0.0f; az = 0.0f; }
    norm3(ax, ay, az);
    // bitangent b = normalize(cross(h, a))
    float bx = hy * az - hz * ay;
    float by = hz * ax - hx * az;
    float bz = hx * ay - hy * ax;
    norm3(bx, by, bz);
    float pd = degen ? 0.0f
                     : atan2f(bx * lx + by * ly + bz * lz,
                              ax * lx + ay * ly + az * lz);

    // half-diff -> continuous indices
    float thdeg = th * RAD2IDX;  // theta_h / (pi/2) * 90
    float idx1 = (th <= 0.0f)
                     ? 0.0f
                     : fminf(fmaxf(sqrtf(fmaxf(thdeg * 90.0f, 0.0f)), 0.0f), 90.0f);
    float idx2 = fminf(fmaxf(td * RAD2IDX, 0.0f), 90.0f);
    if (pd < 0.0f) pd += PI_F;
    float idx3 = fminf(fmaxf(pd * RAD2IDX, 0.0f), 180.0f);

    float f1 = fminf(fmaxf(idx1 - 0.5f, 0.0f), 88.999f);
    float f2 = fminf(fmaxf(idx2 - 0.5f, 0.0f), 88.999f);
    float f3 = fminf(fmaxf(idx3 - 0.5f, 0.0f), 178.999f);
    int i1 = (int)f1, i2 = (int)f2, i3 = (int)f3;
    float d1 = fminf(fmaxf(f1 - (float)i1, 0.0f), 1.0f);
    float d2 = fminf(fmaxf(f2 - (float)i2, 0.0f), 1.0f);
    float d3 = fminf(fmaxf(f3 - (float)i3, 0.0f), 1.0f);
    float e1 = 1.0f - d1, e2 = 1.0f - d2, e3 = 1.0f - d3;

    int base = (i1 * TD_N + i2) * PD_N + i3;
    float r0, r1, r2;
    if (USE_WS) {
        // K constants already folded into the repacked table.
        float4 g000 = tbl[base],          g001 = tbl[base + 1];
        float4 g010 = tbl[base + 180],    g011 = tbl[base + 181];
        float4 g100 = tbl[base + 16200],  g101 = tbl[base + 16201];
        float4 g110 = tbl[base + 16380],  g111 = tbl[base + 16381];
#define TRILERP(c)                                                            \
    fmaxf(e3 * (e1 * (e2 * g000.c + d2 * g010.c) +                            \
                d1 * (e2 * g100.c + d2 * g110.c)) +                           \
              d3 * (e1 * (e2 * g001.c + d2 * g011.c) +                        \
                    d1 * (e2 * g101.c + d2 * g111.c)),                        \
          0.0f)
        r0 = TRILERP(x);
        r1 = TRILERP(y);
        r2 = TRILERP(z);
#undef TRILERP
    } else {
        const int CH = TH_N * TD_N * PD_N;  // 1,458,000
        const float Ks[3] = {1.00f / 1500.0f, 1.15f / 1500.0f, 1.66f / 1500.0f};
        float rv[3];
#pragma unroll
        for (int c = 0; c < 3; ++c) {
            const float* p = brdf + c * CH + base;
            float g000 = p[0],     g001 = p[1];
            float g010 = p[180],   g011 = p[181];
            float g100 = p[16200], g101 = p[16201];
            float g110 = p[16380], g111 = p[16381];
            float A = e2 * g000 + d2 * g010;
            float B = e2 * g100 + d2 * g110;
            float C = e2 * g001 + d2 * g011;
            float D = e2 * g101 + d2 * g111;
            rv[c] = Ks[c] *
                    fmaxf(e3 * (e1 * A + d1 * B) + d3 * (e1 * C + d1 * D), 0.0f);
        }
        r0 = rv[0]; r1 = rv[1]; r2 = rv[2];
    }

    float cosl = fmaxf(lx * nx + ly * ny + lz * nz, 0.0f);
    size_t ob = (size_t)i * 3;
    out[ob]     = r0 * cosl * ir;
    out[ob + 1] = r1 * cosl * ig;
    out[ob + 2] = r2 * cosl * ibl;
}

extern "C" void kernel_launch(void* const* d_in, const int* in_sizes, int n_in,
                              void* d_out, int out_size, void* d_ws, size_t ws_size,
                              hipStream_t stream) {
    const float* brdf   = (const float*)d_in[0];
    const float* normal = (const float*)d_in[1];
    const float* view   = (const float*)d_in[2];
    const float* light  = (const float*)d_in[3];
    const float* inten  = (const float*)d_in[4];
    float* out = (float*)d_out;
    const int n = in_sizes[1] / 3;
    const int texels = TH_N * TD_N * PD_N;
    const bool use_ws = (d_ws != nullptr) && (ws_size >= (size_t)texels * sizeof(float4));
    const int rblocks = (n + KTILE - 1) / KTILE;
    if (use_ws) {
        permute_brdf_kernel<<<(texels + KTILE - 1) / KTILE, KTILE, 0, stream>>>(
            brdf, (float4*)d_ws, texels);
        render_kernel<true><<<rblocks, KTILE, 0, stream>>>(
            brdf, (const float4*)d_ws, normal, view, light, inten, out, n);
    } else {
        render_kernel<false><<<rblocks, KTILE, 0, stream>>>(
            brdf, nullptr, normal, view, light, inten, out, n);
    }
}